// Interaction_GraphConvolution_55963423867450
// MI455X (gfx1250) — compile-verified
//
#include <hip/hip_runtime.h>
#include <hip/hip_bf16.h>

#define IN_F  128
#define OUT_F 128

typedef __attribute__((ext_vector_type(2))) float v2f;
typedef __attribute__((ext_vector_type(8))) float v8f;

// ---------------------------------------------------------------------------
// Kernel 1: wx[m,n] = sum_k X[m,k] * W[n,k] + b[n]   (X: [N,128], W: [128,128])
// One block = 16 rows; 8 waves per block, wave w owns the 16-col tile n0=16w.
// Each wave runs 32 chained V_WMMA_F32_16X16X4_F32 over K=128.
// ---------------------------------------------------------------------------
__global__ __launch_bounds__(256) void gemm_wx_kernel(
    const float* __restrict__ X, const float* __restrict__ W,
    const float* __restrict__ bias, float* __restrict__ wx, int n_nodes)
{
  const int lane  = threadIdx.x & 31;
  const int wavei = threadIdx.x >> 5;            // 0..7 -> column tile
  const int row0  = blockIdx.x << 4;             // 16 rows per block
  const int n0    = wavei << 4;
  const int l15   = lane & 15;
  const int khalf = (lane >> 4) << 1;            // 0 (lanes 0-15) or 2 (lanes 16-31)
  const int n     = n0 + l15;                    // this lane's output column
  const int mbase = row0 + ((lane >> 4) << 3);   // C rows: mbase..mbase+7

  v8f acc = (v8f)0.f;

#if defined(__gfx1250__) && __has_builtin(__builtin_amdgcn_wmma_f32_16x16x4_f32)
  // A 16x4 fragment (ISA 7.12.2): lane l<16 holds A[row0+l, k0+{0,1}],
  // lanes 16-31 hold A[row0+(l-16), k0+{2,3}].
  // B 4x16 fragment: B[k][n] = W[n][k]; N striped across lanes, K split by half-wave.
  const int arow_m = (row0 + l15 < n_nodes) ? (row0 + l15) : (n_nodes - 1); // clamp (N%16==0 here)
  const float* arow = X + (size_t)arow_m * IN_F + khalf;
  const float* brow = W + (size_t)n      * IN_F + khalf;
  #pragma unroll
  for (int k0 = 0; k0 < IN_F; k0 += 4) {
    v2f a, b;
    a.x = arow[k0]; a.y = arow[k0 + 1];
    b.x = brow[k0]; b.y = brow[k0 + 1];
    acc = __builtin_amdgcn_wmma_f32_16x16x4_f32(
        /*neg_a=*/false, a, /*neg_b=*/false, b,
        /*c_mod=*/(short)0, acc, /*reuse_a=*/false, /*reuse_b=*/false);
  }
#else
  // Scalar fallback (host pass / missing builtin) — same output mapping as C layout.
  for (int r = 0; r < 8; ++r) {
    float s = 0.f;
    int mrow = (mbase + r < n_nodes) ? (mbase + r) : (n_nodes - 1);
    for (int k = 0; k < IN_F; ++k)
      s += X[(size_t)mrow * IN_F + k] * W[(size_t)n * IN_F + k];
    acc[r] = s;
  }
#endif

  // C layout: VGPR r -> lanes 0-15: (M=row0+r, N=n0+lane); lanes 16-31: M=row0+8+r.
  const float bv = bias[n];
  #pragma unroll
  for (int r = 0; r < 8; ++r) {
    const int mrow = mbase + r;
    if (mrow < n_nodes) wx[(size_t)mrow * OUT_F + n] = acc[r] + bv;
  }
}

// ---------------------------------------------------------------------------
// Kernel 2: one wave32 per edge. temp[row[e], :] += vals[e]^2 * wx[col[e], :]
// Lane i covers features [4i, 4i+4): float4 gather (L2-resident, 51MB < 192MB)
// + 4 global_atomic_add_f32. Self-loops (row==col) contribute zero -> skip.
// ---------------------------------------------------------------------------
__global__ __launch_bounds__(256) void scatter_kernel(
    const float* __restrict__ wx, const long long* __restrict__ row,
    const long long* __restrict__ col, const float* __restrict__ vals,
    float* __restrict__ temp, int n_edges)
{
  const unsigned tid = blockIdx.x * (unsigned)blockDim.x + threadIdx.x;
  const int e    = (int)(tid >> 5);
  const int lane = threadIdx.x & 31;
  if (e >= n_edges) return;

  const long long r = row[e];
  const long long c = col[e];
  if (r == c) return;

  float v = vals[e];
  v *= v;

  const float4 m = ((const float4*)(wx + (size_t)c * OUT_F))[lane];
  float* dst = temp + (size_t)r * OUT_F + (lane << 2);
  atomicAdd(dst + 0, v * m.x);
  atomicAdd(dst + 1, v * m.y);
  atomicAdd(dst + 2, v * m.z);
  atomicAdd(dst + 3, v * m.w);
}

// ---------------------------------------------------------------------------
// Kernel 3: out = wx * temp (temp was accumulated in d_out), float4 vectorized.
// ---------------------------------------------------------------------------
__global__ __launch_bounds__(256) void hadamard_kernel(
    const float4* __restrict__ wx4, float4* __restrict__ out4, int nq)
{
  const int i = blockIdx.x * blockDim.x + threadIdx.x;
  if (i >= nq) return;
  float4 a = wx4[i];
  float4 t = out4[i];
  a.x *= t.x; a.y *= t.y; a.z *= t.z; a.w *= t.w;
  out4[i] = a;
}

// ---------------------------------------------------------------------------
// Inputs (setup_inputs order):
//   d_in[0] node_features f32 [N*128]
//   d_in[1] edge_index    i64 [2*E]   (row = first E, col = second E)
//   d_in[2] edge_values   f32 [E]
//   d_in[3] W             f32 [128*128]
//   d_in[4] b             f32 [128]
// d_out: f32 [N*128]. d_ws: wx scratch (N*128 f32 = 51.2 MB).
// ---------------------------------------------------------------------------
extern "C" void kernel_launch(void* const* d_in, const int* in_sizes, int n_in,
                              void* d_out, int out_size, void* d_ws, size_t ws_size,
                              hipStream_t stream) {
  const float*     X    = (const float*)d_in[0];
  const long long* ei   = (const long long*)d_in[1];
  const float*     vals = (const float*)d_in[2];
  const float*     W    = (const float*)d_in[3];
  const float*     bias = (const float*)d_in[4];
  float*           out  = (float*)d_out;

  const int n_nodes = in_sizes[0] / IN_F;
  const int n_edges = in_sizes[2];
  const long long* row = ei;
  const long long* col = ei + n_edges;
  float* wx = (float*)d_ws;                       // [n_nodes * OUT_F]

  // 1) wx = X @ W.T + b  (WMMA f32)
  const int row_tiles = (n_nodes + 15) / 16;
  gemm_wx_kernel<<<row_tiles, 256, 0, stream>>>(X, W, bias, wx, n_nodes);

  // 2) temp (in d_out) = 0, then scatter-add messages
  hipMemsetAsync(d_out, 0, (size_t)out_size * sizeof(float), stream);
  const long long sthreads = (long long)n_edges * 32;
  scatter_kernel<<<(int)((sthreads + 255) / 256), 256, 0, stream>>>(
      wx, row, col, vals, out, n_edges);

  // 3) out = wx * temp
  const int nq = out_size / 4;
  hadamard_kernel<<<(nq + 255) / 256, 256, 0, stream>>>(
      (const float4*)wx, (float4*)out, nq);
}